// MTLoss_2619930050995
// MI455X (gfx1250) — compile-verified
//
#include <hip/hip_runtime.h>

#define NA 8732
#define NCLS 21
#define MAXGT 64
#define SEGN 90000
#define NEGR 3
#define THREADS 256
#define NBLK_CLS 1024
#define NBLK_SEG 4096

typedef float v2f __attribute__((ext_vector_type(2)));
typedef float v8f __attribute__((ext_vector_type(8)));
typedef unsigned int u32x4 __attribute__((ext_vector_type(4)));
typedef int i32x8 __attribute__((ext_vector_type(8)));
typedef int i32x4 __attribute__((ext_vector_type(4)));

__device__ __constant__ int c_loff[6] = {0, 5776, 7942, 8542, 8692, 8728};
__device__ __constant__ int c_nbx[6]  = {4, 6, 6, 6, 4, 4};

// Wave-level f32 sum on the matrix pipe: D = ones(16x4) x B(4x16).
// Every D element (m,n) is colsum[n]; d[0] holds rows m=0 / m=8, i.e. two
// copies of the 16 column sums -> 32-lane sum of d[0] == 2 * total.
__device__ __forceinline__ float wave_reduce_f32(float v) {
  v2f a; a[0] = 1.0f; a[1] = 1.0f;
  v2f b; b[0] = v;    b[1] = 0.0f;
  v8f c = {};
  v8f d = __builtin_amdgcn_wmma_f32_16x16x4_f32(false, a, false, b, (short)0, c,
                                                false, false);
  float s = d[0];
  s += __shfl_xor(s, 16, 32);
  s += __shfl_xor(s, 8, 32);
  s += __shfl_xor(s, 4, 32);
  s += __shfl_xor(s, 2, 32);
  s += __shfl_xor(s, 1, 32);
  return 0.5f * s;
}

// Deterministic block reduce; result valid in thread 0. Block = 256 threads.
__device__ __forceinline__ float block_reduce_f32(float v, float* sm8) {
  int lane = threadIdx.x & 31, wid = threadIdx.x >> 5;
  float w = wave_reduce_f32(v);
  if (lane == 0) sm8[wid] = w;
  __syncthreads();
  float r = 0.0f;
  if (threadIdx.x == 0)
    for (int i = 0; i < (int)(blockDim.x >> 5); ++i) r += sm8[i];
  __syncthreads();
  return r;
}

// Deterministic block reduce broadcast to all threads.
__device__ __forceinline__ float block_reduce_bcast_f32(float v, float* sm8,
                                                        float* bc) {
  int lane = threadIdx.x & 31, wid = threadIdx.x >> 5;
  float w = wave_reduce_f32(v);
  if (lane == 0) sm8[wid] = w;
  __syncthreads();
  if (threadIdx.x == 0) {
    float r = 0.0f;
    for (int i = 0; i < (int)(blockDim.x >> 5); ++i) r += sm8[i];
    bc[0] = r;
  }
  __syncthreads();
  float res = bc[0];
  __syncthreads();
  return res;
}

__global__ void k_init(int* writer, int* pos_cnt, int Bn) {
  int n = Bn * NA;
  for (int i = blockIdx.x * blockDim.x + threadIdx.x; i < n;
       i += gridDim.x * blockDim.x)
    writer[i] = -1;
  if (blockIdx.x == 0 && threadIdx.x < 64) pos_cnt[threadIdx.x] = 0;
}

// Last-write-wins scatter: writer[b][flat] = max valid GT index g.
__global__ void k_scatter(const int* idx, int* writer, int Bn) {
  int b = blockIdx.x;
  int g = threadIdx.x;
  if (b >= Bn || g >= MAXGT) return;
  const int* row = idx + (size_t)b * MAXGT * 4;
  bool valid = true;
  for (int j = 0; j <= g; ++j)
    if (row[j * 4 + 1] == -100) { valid = false; break; }
  if (!valid) return;
  int l  = row[g * 4 + 1];
  int fp = row[g * 4 + 2];
  int db = row[g * 4 + 3];
  int lc = min(max(l, 0), 5);
  int flat = c_loff[lc] + fp * c_nbx[lc] + db;
  if (flat >= 0 && flat < NA) atomicMax(&writer[b * NA + flat], g);
}

__global__ void k_cls(const float* __restrict__ Cls, const float* __restrict__ Loc,
                      const float* __restrict__ gt, const float* __restrict__ df,
                      const int* __restrict__ clsb, const int* __restrict__ mining_p,
                      const int* __restrict__ writer, float* __restrict__ cebg,
                      int* pos_cnt, float* cepos_part, float* loc_part,
                      float* negall_part, int Bn) {
  __shared__ float sm8[8];
  int mining = mining_p[0];
  float acc_pos = 0.f, acc_loc = 0.f, acc_na = 0.f;
  int total = Bn * NA;
  for (int i = blockIdx.x * blockDim.x + threadIdx.x; i < total;
       i += gridDim.x * blockDim.x) {
    int b = i / NA;
    const float* x = Cls + (size_t)i * NCLS;
    float xv[NCLS];
    float m = -3.4e38f;
#pragma unroll
    for (int c = 0; c < NCLS; ++c) {
      xv[c] = __builtin_nontemporal_load(x + c);   // streamed once: TH=NT
      m = fmaxf(m, xv[c]);
    }
    float s = 0.f;
#pragma unroll
    for (int c = 0; c < NCLS; ++c) s += expf(xv[c] - m);
    float lse = m + logf(s);
    int w = writer[i];
    float bg = lse - xv[0];
    if (w >= 0) {
      int cl = clsb[b * MAXGT + w];
      cl = min(max(cl, 0), NCLS - 1);
      acc_pos += lse - x[cl];            // reload: no dynamic reg indexing
      atomicAdd(&pos_cnt[b], 1);
      const float* gtp = gt + (size_t)(b * MAXGT + w) * 4;
      const float* dfp = df + (size_t)(b * MAXGT + w) * 4;
      const float* lp  = Loc + (size_t)i * 4;
#pragma unroll
      for (int k = 0; k < 4; ++k) {
        float dl = (gtp[k] - dfp[k]) * 10.0f;      // /LOC_VAR
        float dd = __builtin_nontemporal_load(lp + k) - dl;
        float ad = fabsf(dd);
        acc_loc += (ad < 1.f) ? 0.5f * dd * dd : ad - 0.5f;
      }
      cebg[i] = -1.0f;                   // not a negative candidate
    } else {
      cebg[i] = bg;
      if (!mining) acc_na += bg;         // mining==0: all non-pos are bg
    }
  }
  float r;
  r = block_reduce_f32(acc_pos, sm8); if (threadIdx.x == 0) cepos_part[blockIdx.x] = r;
  r = block_reduce_f32(acc_loc, sm8); if (threadIdx.x == 0) loc_part[blockIdx.x] = r;
  r = block_reduce_f32(acc_na, sm8);  if (threadIdx.x == 0) negall_part[blockIdx.x] = r;
}

// Exact per-image top-k sum of ce_bg. The 8732-float row is pulled into LDS by
// the Tensor Data Mover (one 1-D descriptor per wave, contiguous chunks).
__global__ void k_topk(const float* __restrict__ cebg, const int* pos_cnt,
                       const int* mining_p, float* negtop_part, int Bn) {
  __shared__ __align__(16) float vals[NA];
  __shared__ float sm8[8];
  __shared__ float bc[1];
  int b = blockIdx.x;
  int tid = threadIdx.x;
  int wid = tid >> 5;

  {
    const int chunk = 1092;                       // 7*1092 + 1088 = 8732
    int start = wid * chunk;
    int len = NA - start; if (len > chunk) len = chunk;
    unsigned long long ga =
        (unsigned long long)(size_t)(cebg + (size_t)b * NA + start);
    unsigned lofs = (unsigned)(size_t)(&vals[start]);
    u32x4 g0;
    g0[0] = 1u;                                   // count=1, user descriptor
    g0[1] = lofs;                                 // lds_addr
    g0[2] = (unsigned)(ga & 0xFFFFFFFFu);         // global_addr[31:0]
    g0[3] = (unsigned)((ga >> 32) & 0x01FFFFFFu)  // global_addr[56:32]
            | (2u << 30);                         // type=2 ("image")
    i32x8 g1;
    g1[0] = 0x20000;                              // data_size=4B
    g1[1] = len << 16;                            // tensor_dim0[15:0]
    g1[2] = 1 << 16;                              // dim0 hi=0 | tensor_dim1=1
    g1[3] = len << 16;                            // dim1 hi=0 | tile_dim0=len
    g1[4] = 1;                                    // tile_dim1=1, tile_dim2=0
    g1[5] = len;                                  // tensor_dim0_stride lo32
    g1[6] = 0;
    g1[7] = 0;
    i32x4 gz; gz[0] = 0; gz[1] = 0; gz[2] = 0; gz[3] = 0;
    i32x8 gz8; gz8[0] = 0; gz8[1] = 0; gz8[2] = 0; gz8[3] = 0;
    gz8[4] = 0; gz8[5] = 0; gz8[6] = 0; gz8[7] = 0;
    __builtin_amdgcn_tensor_load_to_lds(g0, g1, gz, gz, gz8, 0);
    __builtin_amdgcn_s_wait_tensorcnt(0);
  }
  __syncthreads();

  int mining = mining_p[0];
  int np = pos_cnt[b];
  int k = min(NEGR * np, NA - np);
  float result = 0.f;
  if (mining && k > 0) {
    unsigned prefix = 0u;
    for (int bit = 31; bit >= 0; --bit) {         // ce_bg >= 0: bits monotone
      unsigned cand = prefix | (1u << bit);
      int cnt = 0;
      for (int i = tid; i < NA; i += THREADS) {
        float v = vals[i];
        if (v >= 0.f && __float_as_uint(v) >= cand) cnt++;
      }
      int tot = (int)block_reduce_bcast_f32((float)cnt, sm8, bc);  // exact
      if (tot >= k) prefix = cand;                // max T: count(>=T) >= k
    }
    float V = __uint_as_float(prefix);            // k-th largest value
    int cgt = 0; float sgt = 0.f;
    for (int i = tid; i < NA; i += THREADS) {
      float v = vals[i];
      if (v >= 0.f && __float_as_uint(v) > prefix) { cgt++; sgt += v; }
    }
    float tc = block_reduce_bcast_f32((float)cgt, sm8, bc);
    float ts = block_reduce_bcast_f32(sgt, sm8, bc);
    if (tid == 0) result = ts + (float)(k - (int)tc) * V;  // tie-exact
  }
  if (tid == 0) negtop_part[b] = result;
}

__global__ void k_seg(const float* __restrict__ Seg, const int* __restrict__ lab,
                      float* seg_part, int Bn) {
  __shared__ float sm8[8];
  float acc = 0.f;
  long total = (long)Bn * SEGN;
  for (long p = (long)blockIdx.x * blockDim.x + threadIdx.x; p < total;
       p += (long)gridDim.x * blockDim.x) {
    int b = (int)(p / SEGN);
    int r = (int)(p - (long)b * SEGN);
    const float* base = Seg + (size_t)b * NCLS * SEGN + r;
    int l = lab[p]; l = min(max(l, 0), NCLS - 1);
    float xv[NCLS];
    float m = -3.4e38f;
#pragma unroll
    for (int c = 0; c < NCLS; ++c) {
      xv[c] = __builtin_nontemporal_load(base + (size_t)c * SEGN);  // 242MB > L2
      m = fmaxf(m, xv[c]);
    }
    float s = 0.f, xl = 0.f;
#pragma unroll
    for (int c = 0; c < NCLS; ++c) {
      s += expf(xv[c] - m);
      xl = (c == l) ? xv[c] : xl;        // no dynamic register indexing
    }
    acc += (m + logf(s)) - xl;
  }
  float r = block_reduce_f32(acc, sm8);
  if (threadIdx.x == 0) seg_part[blockIdx.x] = r;
}

__global__ void k_final(const float* cepos_part, const float* loc_part,
                        const float* negall_part, const float* negtop_part,
                        const float* seg_part, const int* pos_cnt,
                        const int* mining_p, float* out, int Bn) {
  __shared__ __align__(16) float sseg[NBLK_SEG];
  __shared__ float sm8[8];
  int tid = threadIdx.x;

  // Stage seg partials into LDS via async global->LDS copies (ASYNCcnt path).
  for (int i = tid; i < NBLK_SEG / 4; i += THREADS) {
    unsigned lofs = (unsigned)(size_t)(&sseg[i * 4]);
    const float* g = seg_part + (size_t)i * 4;
    asm volatile("global_load_async_to_lds_b128 %0, %1, off"
                 :: "v"(lofs), "v"(g) : "memory");
  }
  asm volatile("s_wait_asynccnt 0" ::: "memory");
  __syncthreads();

  float ce = 0.f, lo = 0.f, na = 0.f, nt = 0.f, sg = 0.f, pc = 0.f;
  for (int i = tid; i < NBLK_CLS; i += THREADS) {
    ce += cepos_part[i]; lo += loc_part[i]; na += negall_part[i];
  }
  for (int i = tid; i < NBLK_SEG; i += THREADS) sg += sseg[i];
  for (int i = tid; i < Bn; i += THREADS) {
    nt += negtop_part[i];
    pc += (float)pos_cnt[i];             // integer-valued: exact in f32
  }
  float r_ce = block_reduce_f32(ce, sm8);
  float r_lo = block_reduce_f32(lo, sm8);
  float r_na = block_reduce_f32(na, sm8);
  float r_nt = block_reduce_f32(nt, sm8);
  float r_sg = block_reduce_f32(sg, sm8);
  float r_pc = block_reduce_f32(pc, sm8);
  if (tid == 0) {
    float numpos = fmaxf(r_pc, 1.0f);
    float neg = mining_p[0] ? r_nt : r_na;
    out[0] = (r_ce + neg) / numpos + r_lo / numpos +
             r_sg / ((float)Bn * (float)SEGN);
  }
}

extern "C" void kernel_launch(void* const* d_in, const int* in_sizes, int n_in,
                              void* d_out, int out_size, void* d_ws, size_t ws_size,
                              hipStream_t stream) {
  (void)n_in; (void)out_size; (void)ws_size;
  const float* Loc = (const float*)d_in[0];
  const float* Cls = (const float*)d_in[1];
  const float* Seg = (const float*)d_in[2];
  const float* gt  = (const float*)d_in[3];
  const float* df  = (const float*)d_in[4];
  const int*   idx = (const int*)d_in[5];
  const int*   clb = (const int*)d_in[6];
  const int*   mining_p = (const int*)d_in[8];
  const int*   slab = (const int*)d_in[9];
  int Bn = in_sizes[6] / MAXGT;   // cls_batch is [B, MAXGT]

  char* ws = (char*)d_ws;
  size_t o = 0;
  int*   writer      = (int*)(ws + o);   o += (size_t)Bn * NA * sizeof(int);
  float* cebg        = (float*)(ws + o); o += (size_t)Bn * NA * sizeof(float);
  int*   pos_cnt     = (int*)(ws + o);   o += 64 * sizeof(int);
  float* cepos_part  = (float*)(ws + o); o += NBLK_CLS * sizeof(float);
  float* loc_part    = (float*)(ws + o); o += NBLK_CLS * sizeof(float);
  float* negall_part = (float*)(ws + o); o += NBLK_CLS * sizeof(float);
  float* negtop_part = (float*)(ws + o); o += 64 * sizeof(float);
  float* seg_part    = (float*)(ws + o); o += NBLK_SEG * sizeof(float);

  int init_blocks = (Bn * NA + THREADS - 1) / THREADS;
  k_init<<<init_blocks, THREADS, 0, stream>>>(writer, pos_cnt, Bn);
  k_scatter<<<Bn, MAXGT, 0, stream>>>(idx, writer, Bn);
  k_cls<<<NBLK_CLS, THREADS, 0, stream>>>(Cls, Loc, gt, df, clb, mining_p, writer,
                                          cebg, pos_cnt, cepos_part, loc_part,
                                          negall_part, Bn);
  k_topk<<<Bn, THREADS, 0, stream>>>(cebg, pos_cnt, mining_p, negtop_part, Bn);
  k_seg<<<NBLK_SEG, THREADS, 0, stream>>>(Seg, slab, seg_part, Bn);
  k_final<<<1, THREADS, 0, stream>>>(cepos_part, loc_part, negall_part,
                                     negtop_part, seg_part, pos_cnt, mining_p,
                                     (float*)d_out, Bn);
}